// CAM_Module_36429912604838
// MI455X (gfx1250) — compile-verified
//
#include <hip/hip_runtime.h>
#include <hip/hip_bf16.h>
#include <cstdint>

// ============================================================================
// CAM (channel-attention) module for MI455X / gfx1250, all-fp32 WMMA.
//
// Fusion:   out = att @ v  where v = Wv@x + bv  ==>  out = (att@Wv)@x + (att@bv)
//           -> v ([8,128,16384] fp32 = 64MB) is never materialized, saving
//              ~128MB of HBM traffic on a bandwidth-bound problem (23.3 TB/s).
//
// Every GEMM uses V_WMMA_F32_16X16X4_F32 (wave32, D = A(16x4) x B(4x16) + C).
// VGPR layouts per CDNA5 ISA 7.12.2:
//   A (16x4 f32, 2 VGPRs): lane L: M = L%16 ; K = 2*(L/16) + vgpr
//   C/D (16x16 f32, 8 VGPRs): vgpr r: M = r + 8*(L/16) ; N = L%16
//   B (4x16 f32, 2 VGPRs): assumed symmetric to A: N = L%16 ; K = 2*(L/16)+vgpr
// ============================================================================

typedef __attribute__((ext_vector_type(2))) float v2f;
typedef __attribute__((ext_vector_type(8))) float v8f;

#define BATCH 8
#define CCH   128
#define HH    128
#define WW    128
#define HWN   (HH * WW)        // 16384
#define PQ    4096             // 64*64 pooled spatial
#define KQK   512              // C * 2 * 2 im2col depth
#define NSPLIT 8               // split-K for energy GEMM

static __device__ __forceinline__ v8f wmma4(v2f a, v2f b, v8f c) {
  // 8 args: (neg_a, A, neg_b, B, c_mod, C, reuse_a, reuse_b)
  return __builtin_amdgcn_wmma_f32_16x16x4_f32(false, a, false, b, (short)0, c,
                                               false, false);
}

static __device__ __forceinline__ float wave_max(float v) {
#pragma unroll
  for (int o = 16; o > 0; o >>= 1) v = fmaxf(v, __shfl_xor(v, o, 32));
  return v;
}
static __device__ __forceinline__ float wave_sum(float v) {
#pragma unroll
  for (int o = 16; o > 0; o >>= 1) v += __shfl_xor(v, o, 32);
  return v;
}

// ---------------------------------------------------------------------------
// Kernel 1: q (WHICH=0) or kT (WHICH=1) via implicit-im2col GEMM.
//   q[b][c][p]  = sum_k W_flat[c][k] * X2[b][k][p] + bias[c]      (row-major)
//   kT[b][p][c] = same, stored transposed for the energy GEMM.
// K index k = chan*4 + dy*2 + dx; lane-half covers dy, vgpr pair covers dx,
// so each B fragment load is ONE coalesced b64 (two consecutive x elements).
// One wave computes a 16(M) x 64(N) strip (one pooled row ph = strip id).
// WHICH is a template param: wave-uniform, so the epilogue is straight-line
// stores with no exec-mask juggling or pointer cndmasks.
// ---------------------------------------------------------------------------
template <int WHICH>
__global__ void cam_qk_kernel(const float* __restrict__ x,
                              const float* __restrict__ Wm,
                              const float* __restrict__ bias,
                              float* __restrict__ dst) {
  int gw = blockIdx.x * (blockDim.x >> 5) + (threadIdx.x >> 5);
  int ns = gw & 63;  gw >>= 6;   // pooled row (ph), 64-wide N strip
  int mt = gw & 7;   gw >>= 3;   // M tile (16 channels)
  int b  = gw;                   // batch

  const float* xb = x + (size_t)b * CCH * HWN;

  const int lane = threadIdx.x & 31;
  const int lh   = lane >> 4;    // lane half -> dy / K-pair select
  const int lm   = lane & 15;
  const int m0   = mt * 16;

  const v8f vz = {0.f, 0.f, 0.f, 0.f, 0.f, 0.f, 0.f, 0.f};
  v8f acc[4] = {vz, vz, vz, vz};

  for (int k0 = 0; k0 < KQK; k0 += 4) {
    const int ich = k0 >> 2;  // input channel
    v2f a = *(const v2f*)(Wm + (size_t)(m0 + lm) * KQK + k0 + 2 * lh);
    // x row for y = 2*ph + dy (dy = lane half)
    const float* xrow = xb + ((size_t)ich * HH + (2 * ns + lh)) * WW;
#pragma unroll
    for (int s = 0; s < 4; ++s) {
      v2f bb = *(const v2f*)(xrow + 2 * (s * 16 + lm));  // dx = 0,1
      acc[s] = wmma4(a, bb, acc[s]);
    }
  }

#pragma unroll
  for (int s = 0; s < 4; ++s) {
#pragma unroll
    for (int r = 0; r < 8; ++r) {
      const int m = m0 + r + 8 * lh;
      const int p = ns * 64 + s * 16 + lm;
      const float val = acc[s][r] + bias[m];
      if (WHICH == 0)
        dst[((size_t)b * CCH + m) * PQ + p] = val;   // q, row-major [c][p]
      else
        dst[((size_t)b * PQ + p) * CCH + m] = val;   // kT, transposed [p][c]
    }
  }
}

// ---------------------------------------------------------------------------
// Kernel 2: energy partials.  epart[b][sp][c][d] = sum_{p in split sp} q[c,p]*k[d,p]
// A = q [128 x 4096] row-major; B = kT [4096 x 128] (coalesced b32 loads).
// ---------------------------------------------------------------------------
__global__ void cam_energy_kernel(const float* __restrict__ q,
                                  const float* __restrict__ kT,
                                  float* __restrict__ epart) {
  int gw = blockIdx.x * (blockDim.x >> 5) + (threadIdx.x >> 5);
  int nt = gw & 7; gw >>= 3;
  int mt = gw & 7; gw >>= 3;
  int sp = gw & 7; gw >>= 3;
  int b  = gw;

  const float* qb = q  + (size_t)b * CCH * PQ;
  const float* kb = kT + (size_t)b * PQ * CCH;

  const int lane = threadIdx.x & 31;
  const int lh = lane >> 4, lm = lane & 15;

  const v8f vz = {0.f, 0.f, 0.f, 0.f, 0.f, 0.f, 0.f, 0.f};
  v8f acc = vz;

  const int kbeg = sp * (PQ / NSPLIT);
  for (int k0 = kbeg; k0 < kbeg + PQ / NSPLIT; k0 += 4) {
    v2f a = *(const v2f*)(qb + (size_t)(mt * 16 + lm) * PQ + k0 + 2 * lh);
    const int kk = k0 + 2 * lh;
    v2f bb;
    bb.x = kb[(size_t)kk * CCH + nt * 16 + lm];
    bb.y = kb[(size_t)(kk + 1) * CCH + nt * 16 + lm];
    acc = wmma4(a, bb, acc);
  }

#pragma unroll
  for (int r = 0; r < 8; ++r) {
    const int c = mt * 16 + r + 8 * lh;
    epart[(((size_t)b * NSPLIT + sp) * CCH + c) * CCH + nt * 16 + lm] = acc[r];
  }
}

// ---------------------------------------------------------------------------
// Kernel 3: reduce split-K partials + softmax(rowmax - energy) per row,
// one wave32 per row (4 columns/lane), plus cb[b][c] = sum_d att*bv[d].
// ---------------------------------------------------------------------------
__global__ void cam_softmax_kernel(const float* __restrict__ epart,
                                   const float* __restrict__ bv,
                                   float* __restrict__ att,
                                   float* __restrict__ cb) {
  int row = blockIdx.x * (blockDim.x >> 5) + (threadIdx.x >> 5);  // 0..1023
  const int b = row >> 7;
  const int c = row & 127;
  const int lane = threadIdx.x & 31;

  float e0 = 0.f, e1 = 0.f, e2 = 0.f, e3 = 0.f;
  for (int sp = 0; sp < NSPLIT; ++sp) {
    const float4 t = *(const float4*)(epart +
        (((size_t)b * NSPLIT + sp) * CCH + c) * CCH + lane * 4);
    e0 += t.x; e1 += t.y; e2 += t.z; e3 += t.w;
  }
  const float m1 = wave_max(fmaxf(fmaxf(e0, e1), fmaxf(e2, e3)));
  const float n0 = m1 - e0, n1 = m1 - e1, n2 = m1 - e2, n3 = m1 - e3;
  const float m2 = wave_max(fmaxf(fmaxf(n0, n1), fmaxf(n2, n3)));
  float p0 = __expf(n0 - m2), p1 = __expf(n1 - m2);
  float p2 = __expf(n2 - m2), p3 = __expf(n3 - m2);
  const float inv = 1.0f / wave_sum(p0 + p1 + p2 + p3);
  p0 *= inv; p1 *= inv; p2 *= inv; p3 *= inv;

  float4 o; o.x = p0; o.y = p1; o.z = p2; o.w = p3;
  *(float4*)(att + ((size_t)b * CCH + c) * CCH + lane * 4) = o;

  const float4 bvv = *(const float4*)(bv + lane * 4);
  float cbp = wave_sum(p0 * bvv.x + p1 * bvv.y + p2 * bvv.z + p3 * bvv.w);
  if (lane == 0) cb[b * CCH + c] = cbp;
}

// ---------------------------------------------------------------------------
// Kernel 4: Mb[b] = att[b] @ Wv   (128x128x128 per batch, WMMA)
// ---------------------------------------------------------------------------
__global__ void cam_mb_kernel(const float* __restrict__ att,
                              const float* __restrict__ Wv,
                              float* __restrict__ Mb) {
  int gw = blockIdx.x * (blockDim.x >> 5) + (threadIdx.x >> 5);
  int nt = gw & 7; gw >>= 3;
  int mt = gw & 7; gw >>= 3;
  int b  = gw;

  const float* ab = att + (size_t)b * CCH * CCH;
  const int lane = threadIdx.x & 31;
  const int lh = lane >> 4, lm = lane & 15;

  const v8f vz = {0.f, 0.f, 0.f, 0.f, 0.f, 0.f, 0.f, 0.f};
  v8f acc = vz;

  for (int k0 = 0; k0 < CCH; k0 += 4) {
    v2f a = *(const v2f*)(ab + (size_t)(mt * 16 + lm) * CCH + k0 + 2 * lh);
    const int kk = k0 + 2 * lh;
    v2f bb;
    bb.x = Wv[(size_t)kk * CCH + nt * 16 + lm];
    bb.y = Wv[(size_t)(kk + 1) * CCH + nt * 16 + lm];
    acc = wmma4(a, bb, acc);
  }
#pragma unroll
  for (int r = 0; r < 8; ++r) {
    Mb[((size_t)b * CCH + mt * 16 + r + 8 * lh) * CCH + nt * 16 + lm] = acc[r];
  }
}

// ---------------------------------------------------------------------------
// Kernel 5: out[b] = Mb[b] @ x[b]  + cb[b] broadcast.  M=128,K=128,N=16384.
// One wave -> 16x64 strip; A fragment reused across 4 N-subtiles.
// ---------------------------------------------------------------------------
__global__ void cam_out_kernel(const float* __restrict__ x,
                               const float* __restrict__ Mb,
                               const float* __restrict__ cb,
                               float* __restrict__ out) {
  int gw = blockIdx.x * (blockDim.x >> 5) + (threadIdx.x >> 5);
  int ns = gw & 255; gw >>= 8;   // 256 strips of 64 over N=16384
  int mt = gw & 7;   gw >>= 3;
  int b  = gw;

  const float* xb  = x  + (size_t)b * CCH * HWN;
  const float* Mbb = Mb + (size_t)b * CCH * CCH;
  const float* cbb = cb + (size_t)b * CCH;

  const int lane = threadIdx.x & 31;
  const int lh = lane >> 4, lm = lane & 15;
  const int m0 = mt * 16;
  const int nbase = ns * 64;

  const v8f vz = {0.f, 0.f, 0.f, 0.f, 0.f, 0.f, 0.f, 0.f};
  v8f acc[4] = {vz, vz, vz, vz};

  for (int k0 = 0; k0 < CCH; k0 += 4) {
    v2f a = *(const v2f*)(Mbb + (size_t)(m0 + lm) * CCH + k0 + 2 * lh);
    const int kk = k0 + 2 * lh;
    const float* r0 = xb + (size_t)kk * HWN + nbase;
    const float* r1 = r0 + HWN;
    if (k0 + 4 < CCH)  // gfx1250 global_prefetch_b8 for the next channel rows
      __builtin_prefetch(r0 + 4 * HWN + lm * 4, 0, 1);
#pragma unroll
    for (int s = 0; s < 4; ++s) {
      v2f bb;
      bb.x = r0[s * 16 + lm];
      bb.y = r1[s * 16 + lm];
      acc[s] = wmma4(a, bb, acc[s]);
    }
  }

#pragma unroll
  for (int s = 0; s < 4; ++s) {
#pragma unroll
    for (int r = 0; r < 8; ++r) {
      const int m = m0 + r + 8 * lh;
      out[((size_t)b * CCH + m) * HWN + nbase + s * 16 + lm] =
          acc[s][r] + cbb[m];
    }
  }
}

// ---------------------------------------------------------------------------
extern "C" void kernel_launch(void* const* d_in, const int* in_sizes, int n_in,
                              void* d_out, int out_size, void* d_ws,
                              size_t ws_size, hipStream_t stream) {
  const float* x  = (const float*)d_in[0];
  const float* Wq = (const float*)d_in[1];
  const float* bq = (const float*)d_in[2];
  const float* Wk = (const float*)d_in[3];
  const float* bk = (const float*)d_in[4];
  const float* Wv = (const float*)d_in[5];
  const float* bv = (const float*)d_in[6];
  float* out = (float*)d_out;

  // Workspace layout (bytes): all regions fully rewritten every call.
  char* ws = (char*)d_ws;
  size_t off = 0;
  float* q     = (float*)(ws + off); off += (size_t)BATCH * CCH * PQ * 4;      // 16 MB
  float* kT    = (float*)(ws + off); off += (size_t)BATCH * PQ * CCH * 4;      // 16 MB
  float* epart = (float*)(ws + off); off += (size_t)BATCH * NSPLIT * CCH * CCH * 4; // 4 MB
  float* att   = (float*)(ws + off); off += (size_t)BATCH * CCH * CCH * 4;     // 512 KB
  float* Mb    = (float*)(ws + off); off += (size_t)BATCH * CCH * CCH * 4;     // 512 KB
  float* cb    = (float*)(ws + off); off += (size_t)BATCH * CCH * 4;           // 4 KB
  (void)off; (void)ws_size; (void)in_sizes; (void)n_in; (void)out_size;

  const int TPB = 256;                 // 8 wave32 per block
  const int WPB = TPB >> 5;

  // 1) q and kT : 8 * 8 * 64 = 4096 waves each (uniform template instances)
  cam_qk_kernel<0><<<4096 / WPB, TPB, 0, stream>>>(x, Wq, bq, q);
  cam_qk_kernel<1><<<4096 / WPB, TPB, 0, stream>>>(x, Wk, bk, kT);
  // 2) energy split-K partials : 8 * 8 * 8 * 8 = 4096 waves
  cam_energy_kernel<<<4096 / WPB, TPB, 0, stream>>>(q, kT, epart);
  // 3) softmax rows : 8 * 128 = 1024 waves
  cam_softmax_kernel<<<1024 / WPB, TPB, 0, stream>>>(epart, bv, att, cb);
  // 4) Mb = att @ Wv : 8 * 64 = 512 waves
  cam_mb_kernel<<<512 / WPB, TPB, 0, stream>>>(att, Wv, Mb);
  // 5) out = Mb @ x + cb : 8 * 8 * 256 = 16384 waves
  cam_out_kernel<<<16384 / WPB, TPB, 0, stream>>>(x, Mb, cb, out);
}